// NeigborContrast_73598559584742
// MI455X (gfx1250) — compile-verified
//
#include <hip/hip_runtime.h>
#include <hip/hip_bf16.h>
#include <math.h>

// ---------------------------------------------------------------------------
// Problem constants (from reference)
// ---------------------------------------------------------------------------
#define BSZ   8
#define NN    10000
#define DD    256
#define EE    160000
#define KK    5
#define NEG_INF_F (-1000000000.0f)
#define TOT_LOGITS (2.0f * BSZ * NN)

typedef __attribute__((ext_vector_type(2))) float v2f;
typedef __attribute__((ext_vector_type(8))) float v8f;

// ---------------------------------------------------------------------------
// 1) Count edges per row
// ---------------------------------------------------------------------------
__global__ void count_edges_kernel(const long long* __restrict__ ei,
                                   int* __restrict__ counts) {
  int e = blockIdx.x * blockDim.x + threadIdx.x;
  if (e < EE) {
    int row = (int)ei[e];            // edge_index[0][e]
    atomicAdd(&counts[row], 1);
  }
}

// ---------------------------------------------------------------------------
// 2) Exclusive scan over N row counts (single block, 1024 threads)
// ---------------------------------------------------------------------------
__global__ void scan_counts_kernel(const int* __restrict__ counts,
                                   int* __restrict__ offsets,
                                   int* __restrict__ cursor) {
  __shared__ int ssum[1024];
  const int tid = threadIdx.x;
  constexpr int CH = (NN + 1023) / 1024;   // 10
  int local[CH];
  int base = tid * CH;
  int s = 0;
  for (int i = 0; i < CH; ++i) {
    int c = (base + i < NN) ? counts[base + i] : 0;
    local[i] = s;
    s += c;
  }
  ssum[tid] = s;
  __syncthreads();
  // Hillis-Steele inclusive scan
  for (int off = 1; off < 1024; off <<= 1) {
    int v = (tid >= off) ? ssum[tid - off] : 0;
    __syncthreads();
    ssum[tid] += v;
    __syncthreads();
  }
  int pre = (tid == 0) ? 0 : ssum[tid - 1];
  for (int i = 0; i < CH; ++i) {
    if (base + i < NN) {
      offsets[base + i] = pre + local[i];
      cursor[base + i]  = pre + local[i];
    }
  }
}

// ---------------------------------------------------------------------------
// 3) Scatter edge ids into per-row buckets
// ---------------------------------------------------------------------------
__global__ void scatter_edges_kernel(const long long* __restrict__ ei,
                                     int* __restrict__ cursor,
                                     int* __restrict__ bucket) {
  int e = blockIdx.x * blockDim.x + threadIdx.x;
  if (e < EE) {
    int row = (int)ei[e];
    int pos = atomicAdd(&cursor[row], 1);
    bucket[pos] = e;
  }
}

// ---------------------------------------------------------------------------
// 4) Exact per-row top-5 (value desc, col asc; diag = 1.0; dup edges: last wins)
// ---------------------------------------------------------------------------
__device__ __forceinline__ void top5_insert(float w, int col, float* bw, int* bi) {
  // keep sorted: bw[0] >= bw[1] >= ... ; tie -> smaller col first
  for (int k = 0; k < KK; ++k) {
    bool better = (w > bw[k]) || (w == bw[k] && col < bi[k]);
    if (better) {
      for (int j = KK - 1; j > k; --j) { bw[j] = bw[j - 1]; bi[j] = bi[j - 1]; }
      bw[k] = w; bi[k] = col;
      return;
    }
  }
}

__global__ void topk_rows_kernel(const long long* __restrict__ ei,
                                 const float* __restrict__ ew,
                                 const int* __restrict__ offsets,
                                 const int* __restrict__ counts,
                                 const int* __restrict__ bucket,
                                 int* __restrict__ tkIdx,
                                 float* __restrict__ tkW) {
  int row = blockIdx.x * blockDim.x + threadIdx.x;
  if (row >= NN) return;
  int beg = offsets[row];
  int len = counts[row];
  float bw[KK];
  int   bi[KK];
  for (int k = 0; k < KK; ++k) { bw[k] = -1.0f; bi[k] = 0x7fffffff; }
  // diagonal entry: weight 1.0 at col=row (uniform weights are < 1.0)
  top5_insert(1.0f, row, bw, bi);
  for (int t = 0; t < len; ++t) {
    int e   = bucket[beg + t];
    int col = (int)ei[EE + e];        // edge_index[1][e]
    // dense scatter semantics: last edge writing (row,col) wins
    bool keep = true;
    for (int u = 0; u < len; ++u) {
      int e2 = bucket[beg + u];
      if (e2 > e && (int)ei[EE + e2] == col) { keep = false; break; }
    }
    if (!keep) continue;
    if (col == row) {
      // diagonal overwritten to 1.0 by reference after edge scatter -> skip
      continue;
    }
    top5_insert(ew[e], col, bw, bi);
  }
  for (int k = 0; k < KK; ++k) {
    bool filled = (bw[k] > 0.0f);
    // unfilled slots correspond to zero entries in the dense matrix; the
    // reference masks them (valid = w > 0), their index is irrelevant (attn->0)
    tkW[row * KK + k]   = filled ? bw[k] : 0.0f;
    tkIdx[row * KK + k] = filled ? bi[k] : 0;
  }
}

// ---------------------------------------------------------------------------
// 5) Fused gather + masked softmax attention + aggregate. One wave32 per (b,n).
//    Lane owns 8 contiguous d-elements (256 / 32) -> 2x float4 (b128) loads.
//    This is the roofline-dominant stage: ~410 MB of gathered z1 reads.
// ---------------------------------------------------------------------------
__global__ void attn_agg_kernel(const float* __restrict__ z1,
                                const int* __restrict__ tkIdx,
                                const float* __restrict__ tkW,
                                const float* __restrict__ sa_w,
                                const float* __restrict__ sa_b_p,
                                float* __restrict__ agg) {
  int gw   = (blockIdx.x * blockDim.x + threadIdx.x) >> 5;
  int lane = threadIdx.x & 31;
  if (gw >= BSZ * NN) return;
  int b = gw / NN;
  int n = gw - b * NN;
  const float sa_b = sa_b_p[0];

  float4 sw0, sw1;
  {
    const float4* p = (const float4*)(sa_w + lane * 8);
    sw0 = p[0]; sw1 = p[1];
  }

  const float* zb = z1 + (size_t)b * NN * DD;
  float4 v0[KK], v1[KK];
  float logit[KK];
  #pragma unroll
  for (int k = 0; k < KK; ++k) {
    int   idx = tkIdx[n * KK + k];
    float w   = tkW[n * KK + k];
    const float4* row = (const float4*)(zb + (size_t)idx * DD + lane * 8);
    v0[k] = row[0];
    v1[k] = row[1];
    float p = v0[k].x * sw0.x + v0[k].y * sw0.y + v0[k].z * sw0.z + v0[k].w * sw0.w
            + v1[k].x * sw1.x + v1[k].y * sw1.y + v1[k].z * sw1.z + v1[k].w * sw1.w;
    // wave32 butterfly reduce -> every lane holds the full dot product
    for (int off = 16; off > 0; off >>= 1) p += __shfl_xor(p, off);
    logit[k] = (w > 0.0f) ? (p + sa_b) : NEG_INF_F;
  }
  // softmax over K=5 (replicated in every lane)
  float m = logit[0];
  #pragma unroll
  for (int k = 1; k < KK; ++k) m = fmaxf(m, logit[k]);
  float den = 0.0f;
  float ex[KK];
  #pragma unroll
  for (int k = 0; k < KK; ++k) { ex[k] = expf(logit[k] - m); den += ex[k]; }
  float inv = 1.0f / den;

  float4 a0 = make_float4(0.f, 0.f, 0.f, 0.f);
  float4 a1 = make_float4(0.f, 0.f, 0.f, 0.f);
  #pragma unroll
  for (int k = 0; k < KK; ++k) {
    float w = ex[k] * inv;
    a0.x += w * v0[k].x; a0.y += w * v0[k].y; a0.z += w * v0[k].z; a0.w += w * v0[k].w;
    a1.x += w * v1[k].x; a1.y += w * v1[k].y; a1.z += w * v1[k].z; a1.w += w * v1[k].w;
  }
  float4* out = (float4*)(agg + (size_t)gw * DD + lane * 8);
  out[0] = a0;
  out[1] = a1;
}

// ---------------------------------------------------------------------------
// 6) H = agg @ lin_w^T + lin_b via V_WMMA_F32_16X16X4_F32 (exact f32).
//    One wave per 16x16 output tile; K=256 -> 64 WMMA steps (4x unrolled).
//    h[m,n] = sum_k agg[m,k] * W[n,k]  (W row-major [n][k] -> contiguous b64
//    loads per lane). W is 256 KB and stays resident in the 192 MB L2, so the
//    16x A/B tile re-read amplification never reaches HBM.
// ---------------------------------------------------------------------------
__global__ void gemm_wmma_kernel(const float* __restrict__ A,   // M x 256
                                 const float* __restrict__ W,   // 256 x 256
                                 const float* __restrict__ bias,// 256
                                 float* __restrict__ H,         // M x 256
                                 int M) {
  const int wave = threadIdx.x >> 5;         // 8 waves / block
  const int lane = threadIdx.x & 31;
  int tileId = blockIdx.x * 8 + wave;        // total tiles = (M/16) * 16
  int mTile  = tileId >> 4;
  int nTile  = tileId & 15;
  if (mTile * 16 >= M) return;               // wave-uniform; EXEC stays full
  const int mBase   = mTile * 16;
  const int nBase   = nTile * 16;
  const int half    = lane >> 4;             // 0: K pair {0,1}; 1: K pair {2,3}
  const int lanelo  = lane & 15;

  const float* aRow = A + (size_t)(mBase + lanelo) * DD;   // A[M=lanelo][*]
  const float* wRow = W + (size_t)(nBase + lanelo) * DD;   // B[K][N=lanelo] = W[N][K]
  v8f acc = {};
  #pragma unroll 4
  for (int k0 = 0; k0 < DD; k0 += 4) {
    int k = k0 + half * 2;
    v2f a = { aRow[k], aRow[k + 1] };
    v2f b = { wRow[k], wRow[k + 1] };
    acc = __builtin_amdgcn_wmma_f32_16x16x4_f32(
        /*neg_a=*/false, a, /*neg_b=*/false, b,
        /*c_mod=*/(short)0, acc, /*reuse_a=*/false, /*reuse_b=*/false);
  }
  // C/D layout: VGPR r -> lanes 0-15: M=mBase+r ; lanes 16-31: M=mBase+8+r ;
  //             N = nBase + (lane & 15)
  int   n  = nBase + lanelo;
  float bn = bias[n];
  #pragma unroll
  for (int r = 0; r < 8; ++r) {
    int m = mBase + half * 8 + r;
    H[(size_t)m * DD + n] = acc[r] + bn;
  }
}

// ---------------------------------------------------------------------------
// 7) Cosine score + BCE loss/accuracy. One wave32 per (b,n).
//    fake=0: pair (z2[b,n], h[b,n]),         label 1
//    fake=1: pair (z2[b,n], h[pb(b),pn(n)]), label 0
//    NOTE: the reference permutations come from JAX threefry (host RNG) which
//    cannot be reproduced on-device; deterministic coprime affine permutations
//    are used as data-independent stand-ins.
// ---------------------------------------------------------------------------
__device__ __forceinline__ int perm_b(int b) { return (5 * b + 3) & 7; }  // gcd(5,8)=1
__device__ __forceinline__ int perm_n(int n) {
  return (int)(((long long)7919 * n + 4231) % NN);                        // gcd(7919,10000)=1
}

__global__ void score_loss_kernel(const float* __restrict__ z2,
                                  const float* __restrict__ h,
                                  float* __restrict__ out,   // out[0]=loss, out[1]=acc
                                  int fake) {
  int gw   = (blockIdx.x * blockDim.x + threadIdx.x) >> 5;
  int lane = threadIdx.x & 31;
  if (gw >= BSZ * NN) return;
  int b = gw / NN;
  int n = gw - b * NN;

  const float4* s = (const float4*)(z2 + (size_t)gw * DD + lane * 8);
  const float4* hp;
  if (fake) {
    hp = (const float4*)(h + ((size_t)perm_b(b) * NN + perm_n(n)) * DD + lane * 8);
  } else {
    hp = (const float4*)(h + (size_t)gw * DD + lane * 8);
  }
  float4 s0 = s[0], s1 = s[1];
  float4 h0 = hp[0], h1 = hp[1];
  float ss = s0.x*s0.x + s0.y*s0.y + s0.z*s0.z + s0.w*s0.w
           + s1.x*s1.x + s1.y*s1.y + s1.z*s1.z + s1.w*s1.w;
  float hh = h0.x*h0.x + h0.y*h0.y + h0.z*h0.z + h0.w*h0.w
           + h1.x*h1.x + h1.y*h1.y + h1.z*h1.z + h1.w*h1.w;
  float sh = s0.x*h0.x + s0.y*h0.y + s0.z*h0.z + s0.w*h0.w
           + s1.x*h1.x + s1.y*h1.y + s1.z*h1.z + s1.w*h1.w;
  for (int off = 16; off > 0; off >>= 1) {
    ss += __shfl_xor(ss, off);
    hh += __shfl_xor(hh, off);
    sh += __shfl_xor(sh, off);
  }
  if (lane == 0) {
    float ns = fmaxf(sqrtf(ss), 1e-12f);
    float nh = fmaxf(sqrtf(hh), 1e-12f);
    float l  = sh / (ns * nh);
    float lbl = fake ? 0.0f : 1.0f;
    float loss = fmaxf(l, 0.0f) - l * lbl + log1pf(expf(-fabsf(l)));
    // sigmoid(l) > 0.5  <=>  l > 0
    float pred = (l > 0.0f) ? 1.0f : 0.0f;
    float acc  = (pred == lbl) ? 1.0f : 0.0f;
    atomicAdd(&out[0], loss / TOT_LOGITS);
    atomicAdd(&out[1], acc  / TOT_LOGITS);
  }
}

// ---------------------------------------------------------------------------
// Host-side orchestration
// ---------------------------------------------------------------------------
extern "C" void kernel_launch(void* const* d_in, const int* in_sizes, int n_in,
                              void* d_out, int out_size, void* d_ws, size_t ws_size,
                              hipStream_t stream) {
  const float*     z1  = (const float*)d_in[0];
  const float*     z2  = (const float*)d_in[1];
  const long long* ei  = (const long long*)d_in[2];   // edge_index (2, E) int64
  const float*     ew  = (const float*)d_in[3];
  const float*     saw = (const float*)d_in[4];
  const float*     sab = (const float*)d_in[5];       // scalar
  const float*     lw  = (const float*)d_in[6];
  const float*     lb  = (const float*)d_in[7];
  float*           out = (float*)d_out;

  // workspace layout (256B aligned slices)
  size_t off = 0;
  auto take = [&](size_t bytes) -> char* {
    char* p = (char*)d_ws + off;
    off = (off + bytes + 255) & ~(size_t)255;
    return p;
  };
  int*   counts  = (int*)  take(NN * sizeof(int));
  int*   offsets = (int*)  take(NN * sizeof(int));
  int*   cursor  = (int*)  take(NN * sizeof(int));
  int*   bucket  = (int*)  take(EE * sizeof(int));
  int*   tkIdx   = (int*)  take(NN * KK * sizeof(int));
  float* tkW     = (float*)take(NN * KK * sizeof(float));
  float* agg     = (float*)take((size_t)BSZ * NN * DD * sizeof(float));
  float* hbuf    = (float*)take((size_t)BSZ * NN * DD * sizeof(float));
  if (off > ws_size) return;  // insufficient scratch (deterministic no-op)

  hipMemsetAsync(counts, 0, NN * sizeof(int), stream);
  hipMemsetAsync(out, 0, 2 * sizeof(float), stream);

  count_edges_kernel<<<(EE + 255) / 256, 256, 0, stream>>>(ei, counts);
  scan_counts_kernel<<<1, 1024, 0, stream>>>(counts, offsets, cursor);
  scatter_edges_kernel<<<(EE + 255) / 256, 256, 0, stream>>>(ei, cursor, bucket);
  topk_rows_kernel<<<(NN + 255) / 256, 256, 0, stream>>>(ei, ew, offsets, counts,
                                                         bucket, tkIdx, tkW);

  // 80000 (b,n) pairs, 8 waves per 256-thread block -> 10000 blocks
  attn_agg_kernel<<<(BSZ * NN) / 8, 256, 0, stream>>>(z1, tkIdx, tkW, saw, sab, agg);

  // tiles = (M/16) m-tiles * 16 n-tiles = 80000; 8 waves/block -> 10000 blocks
  const int M = BSZ * NN;
  const int gemmBlocks = ((M / 16) * 16) / 8;   // 10000
  gemm_wmma_kernel<<<gemmBlocks, 256, 0, stream>>>(agg, lw, lb, hbuf, M);

  score_loss_kernel<<<(BSZ * NN) / 8, 256, 0, stream>>>(z2, hbuf, out, 0);
  score_loss_kernel<<<(BSZ * NN) / 8, 256, 0, stream>>>(z2, hbuf, out, 1);
}